// AdaptiveMixedCoding_41077067219221
// MI455X (gfx1250) — compile-verified
//
#include <hip/hip_runtime.h>
#include <math.h>

// MI455X / gfx1250, wave32. fp32 WMMA 16x16x4 + async global->LDS staging.

typedef __attribute__((ext_vector_type(2))) float v2f;
typedef __attribute__((ext_vector_type(8))) float v8f;

#define EPSF     1e-8f
#define MASK_VAL -1.0f
#define INV_T    10.0f   // 1/TEMPERATURE
#define KC       32      // K-chunk for the streaming GEMM
#define STA      36      // padded row stride of streaming buffers (bank stride 36%64 -> conflict-free frags)
#define STG      68      // padded row stride of G / cc-kernel tiles

// ---- wave32 reductions -----------------------------------------------------
__device__ __forceinline__ float wredsum(float v) {
  #pragma unroll
  for (int o = 16; o; o >>= 1) v += __shfl_xor(v, o, 32);
  return v;
}
__device__ __forceinline__ float wredmax(float v) {
  #pragma unroll
  for (int o = 16; o; o >>= 1) v = fmaxf(v, __shfl_xor(v, o, 32));
  return v;
}

// ---- WMMA f32 16x16x4 fragment load from padded LDS tile -------------------
// A-layout (16x4, MxK): lanes 0-15 hold M=lane, K={kk,kk+1}; lanes 16-31 K={kk+2,kk+3}.
// B-layout (4x16, KxN) mirrors it with N=lane&15.
__device__ __forceinline__ v2f ldsfrag(const float* s, int stride, int rbase, int kk, int lane) {
  int lm = lane & 15;
  int kh = (lane >> 4) << 1;
  const float* p = s + (rbase + lm) * stride + kk + kh;
  v2f f; f[0] = p[0]; f[1] = p[1];
  return f;
}

__device__ __forceinline__ v8f wmma_f32(v2f a, v2f b, v8f c) {
  return __builtin_amdgcn_wmma_f32_16x16x4_f32(false, a, false, b, (short)0, c,
                                               false, false);
}

// ---- CDNA5 async global -> LDS (ASYNCcnt-tracked, VGPR-free staging) -------
// GVS mode: global address is the 64-bit SGPR base plus the 32-bit VGPR byte offset; LDS address comes from the VDST VGPR
__device__ __forceinline__ void async_b128(unsigned lds_off, const float* sbase,
                                           unsigned voff_bytes) {
  asm volatile("global_load_async_to_lds_b128 %0, %1, %2"
               :: "v"(lds_off), "v"(voff_bytes), "s"(sbase) : "memory");
}
__device__ __forceinline__ void wait_async0() {
  asm volatile("s_wait_asynccnt 0" ::: "memory");
}

// ============================================================================
// Kernel 1: row L2 norms for imgs (64*36 rows) and caps (64*50 rows), D=1024.
// ============================================================================
__global__ __launch_bounds__(128) void norms_kernel(const float* __restrict__ imgs,
                                                    const float* __restrict__ caps,
                                                    float* __restrict__ ni,
                                                    float* __restrict__ nc) {
  int wave = (blockIdx.x * 128 + threadIdx.x) >> 5;
  int lane = threadIdx.x & 31;
  const int NI = 64 * 36, NC = 64 * 50;
  if (wave >= NI + NC) return;
  const float* src = (wave < NI) ? (imgs + (size_t)wave * 1024)
                                 : (caps + (size_t)(wave - NI) * 1024);
  float s = 0.f;
  for (int k = lane; k < 1024; k += 32) { float v = src[k]; s += v * v; }
  s = wredsum(s);
  if (lane == 0) {
    float n = sqrtf(s);
    if (wave < NI) ni[wave] = n; else nc[wave - NI] = n;
  }
}

// ============================================================================
// Kernel 2: CC[c] = caps[c] @ caps[c]^T  -> [64, 50, 50], via fp32 WMMA.
// ============================================================================
__global__ __launch_bounds__(256) void cc_kernel(const float* __restrict__ caps,
                                                 float* __restrict__ cc) {
  __shared__ float sC[64 * STG];
  int c = blockIdx.x;
  int tid = threadIdx.x, lane = tid & 31, wave = tid >> 5;
  int t0 = wave, t1 = wave + 8;            // tile = mt*4 + nt
  v8f acc0 = {}; v8f acc1 = {};
  for (int kc = 0; kc < 1024; kc += 64) {
    __syncthreads();
    for (int idx = tid; idx < 64 * 16; idx += 256) {
      int row = idx >> 4, q = idx & 15;
      float4 v = {0.f, 0.f, 0.f, 0.f};
      if (row < 50)
        v = *(const float4*)(caps + ((size_t)c * 50 + row) * 1024 + kc + q * 4);
      *(float4*)(sC + row * STG + q * 4) = v;
    }
    __syncthreads();
    #pragma unroll 4
    for (int kk = 0; kk < 64; kk += 4) {
      v2f a0 = ldsfrag(sC, STG, (t0 >> 2) * 16, kk, lane);
      v2f b0 = ldsfrag(sC, STG, (t0 & 3) * 16, kk, lane);
      acc0 = wmma_f32(a0, b0, acc0);
      v2f a1 = ldsfrag(sC, STG, (t1 >> 2) * 16, kk, lane);
      v2f b1 = ldsfrag(sC, STG, (t1 & 3) * 16, kk, lane);
      acc1 = wmma_f32(a1, b1, acc1);
    }
  }
  int lm = lane & 15, hb = (lane >> 4) * 8;
  #pragma unroll
  for (int j = 0; j < 8; j++) {
    int m0 = (t0 >> 2) * 16 + hb + j, n0 = (t0 & 3) * 16 + lm;
    if (m0 < 50 && n0 < 50) cc[(size_t)c * 2500 + m0 * 50 + n0] = acc0[j];
    int m1 = (t1 >> 2) * 16 + hb + j, n1 = (t1 & 3) * 16 + lm;
    if (m1 < 50 && n1 < 50) cc[(size_t)c * 2500 + m1 * 50 + n1] = acc1[j];
  }
}

// ============================================================================
// Kernel 3: one WG per (i,c). Phase 1: G = imgs[i] @ caps[c]^T (36x50, K=1024)
// fp32 WMMA; K-chunks double-buffered in LDS via async global->LDS loads.
// 192 threads = 6 waves; wave owns 1 M-tile x 2 N-tiles (A-frag reuse).
// Phase 2: masked argmax + softmax over W, mixed attention A,
// num = A.G row-dot, denom^2 = A CC A^T.
// ============================================================================
__global__ __launch_bounds__(192) void fused_kernel(const float* __restrict__ imgs,
                                                    const float* __restrict__ caps,
                                                    const int* __restrict__ img_lens,
                                                    const int* __restrict__ cap_lens,
                                                    const float* __restrict__ alpha_p,
                                                    const float* __restrict__ ni,
                                                    const float* __restrict__ nc,
                                                    const float* __restrict__ cc,
                                                    float* __restrict__ out) {
  __shared__ float sA[2][48 * STA];      // imgs chunk (double buffered), rows 36..47 zero
  __shared__ float sB[2][64 * STA];      // caps chunk (double buffered), rows 50..63 zero
  __shared__ float sG[48 * STG];         // raw dot products G[r][w]
  __shared__ float sCC[2500];            // caps Gram for this c
  __shared__ float sAttn[6][64];         // mixed attention per active row

  int i = blockIdx.x >> 6;
  int c = blockIdx.x & 63;
  int tid = threadIdx.x, lane = tid & 31, wave = tid >> 5;
  int mt  = wave >> 1;                   // 0..2 : M tile
  int ntb = (wave & 1) * 2;              // 0 or 2: first of two N tiles

  const float* gA = imgs + (size_t)i * 36 * 1024;
  const float* gB = caps + (size_t)c * 50 * 1024;
  unsigned aBase[2] = { (unsigned)(uintptr_t)&sA[0][0], (unsigned)(uintptr_t)&sA[1][0] };
  unsigned bBase[2] = { (unsigned)(uintptr_t)&sB[0][0], (unsigned)(uintptr_t)&sB[1][0] };

  // caption Gram into LDS
  for (int idx = tid; idx < 2500; idx += 192)
    sCC[idx] = cc[(size_t)c * 2500 + idx];
  // zero the padding rows once (async engine never touches them)
  for (int idx = tid; idx < 2 * 12 * STA; idx += 192) {
    int b = idx / (12 * STA); sA[b][36 * STA + idx % (12 * STA)] = 0.f;
  }
  for (int idx = tid; idx < 2 * 14 * STA; idx += 192) {
    int b = idx / (14 * STA); sB[b][50 * STA + idx % (14 * STA)] = 0.f;
  }

  // one chunk = (36 rows A + 50 rows B) x 32 floats, as lane-level B128 moves
  auto issue_chunk = [&](int t, int buf) {
    int kc = t * KC;
    for (int idx = tid; idx < 288 + 400; idx += 192) {
      if (idx < 288) {                         // A: 36 rows x 8 float4
        int row = idx >> 3, q = idx & 7;
        async_b128(aBase[buf] + (unsigned)((row * STA + q * 4) * 4), gA,
                   (unsigned)((row * 1024 + kc + q * 4) * 4));
      } else {                                 // B: 50 rows x 8 float4
        int j = idx - 288, row = j >> 3, q = j & 7;
        async_b128(bBase[buf] + (unsigned)((row * STA + q * 4) * 4), gB,
                   (unsigned)((row * 1024 + kc + q * 4) * 4));
      }
    }
  };

  v8f acc0 = {}; v8f acc1 = {};
  issue_chunk(0, 0);
  wait_async0();
  __syncthreads();                             // pads zeroed + chunk 0 resident

  for (int t = 0; t < 1024 / KC; ++t) {
    int cur = t & 1;
    if (t + 1 < 1024 / KC) issue_chunk(t + 1, cur ^ 1);   // overlap next chunk
    const float* pa = sA[cur];
    const float* pb = sB[cur];
    #pragma unroll
    for (int kk = 0; kk < KC; kk += 4) {
      v2f a  = ldsfrag(pa, STA, mt * 16, kk, lane);
      v2f b0 = ldsfrag(pb, STA, ntb * 16, kk, lane);
      v2f b1 = ldsfrag(pb, STA, ntb * 16 + 16, kk, lane);
      acc0 = wmma_f32(a, b0, acc0);
      acc1 = wmma_f32(a, b1, acc1);
    }
    wait_async0();                             // next chunk landed (this wave)
    __syncthreads();                           // ... and for every wave
  }

  {
    // C-layout: VGPR j -> (M = base + j + (lane<16?0:8), N = base + (lane&15))
    int lm = lane & 15, hb = (lane >> 4) * 8;
    #pragma unroll
    for (int j = 0; j < 8; j++) {
      sG[(mt * 16 + hb + j) * STG + ntb * 16 + lm]      = acc0[j];
      sG[(mt * 16 + hb + j) * STG + ntb * 16 + 16 + lm] = acc1[j];
    }
  }
  __syncthreads();

  int img_len = img_lens[i];
  int cap_len = cap_lens[c];
  float a_mix = 1.f / (1.f + expf(-alpha_p[0]));   // sigmoid(alpha)

  // Phase 2: wave per image row r; each lane owns w = lane and w = lane+32.
  for (int r = wave; r < 36; r += 6) {
    float res = MASK_VAL;
    if (r < img_len) {
      float niv = ni[i * 36 + r] + EPSF;
      int w0 = lane, w1 = lane + 32;
      bool v0 = w0 < cap_len;                       // w0 < 32 < 50 always
      bool v1 = (w1 < 50) && (w1 < cap_len);
      float g0 = sG[r * STG + w0];
      float g1 = (w1 < 50) ? sG[r * STG + w1] : 0.f;
      float S0 = v0 ? g0 / (niv * (nc[c * 50 + w0] + EPSF)) : 0.f;
      float S1 = v1 ? g1 / (niv * (nc[c * 50 + w1] + EPSF)) : 0.f;

      // hard argmax over where(mask, S, -1); first occurrence wins ties
      float h0 = v0 ? S0 : -1.0f;
      float h1 = v1 ? S1 : ((w1 < 50) ? -1.0f : -3e38f);
      float bv = (h1 > h0) ? h1 : h0;
      int   bi = (h1 > h0) ? w1 : w0;
      #pragma unroll
      for (int o = 16; o; o >>= 1) {
        float ov = __shfl_xor(bv, o, 32);
        int   oi = __shfl_xor(bi, o, 32);
        if (ov > bv || (ov == bv && oi < bi)) { bv = ov; bi = oi; }
      }

      // masked softmax over S/T (max-shifted; invalid -> exactly 0)
      float mx = wredmax(fmaxf(v0 ? S0 : -3e38f, v1 ? S1 : -3e38f));
      float e0 = v0 ? expf((S0 - mx) * INV_T) : 0.f;
      float e1 = v1 ? expf((S1 - mx) * INV_T) : 0.f;
      float inv = 1.f / wredsum(e0 + e1);

      float A0 = v0 ? (a_mix * e0 * inv + (1.f - a_mix) * ((w0 == bi) ? 1.f : 0.f)) : 0.f;
      float A1 = v1 ? (a_mix * e1 * inv + (1.f - a_mix) * ((w1 == bi) ? 1.f : 0.f)) : 0.f;
      sAttn[wave][w0] = A0;
      sAttn[wave][w1] = A1;

      // num = sum_w A[w] * (caps_w . imgs_r) = sum_w A[w] * G[r,w]
      float num = wredsum(A0 * g0 + A1 * g1);

      // same-wave LDS RAW: attention writes must land before the gathers below
      asm volatile("s_wait_dscnt 0" ::: "memory");

      // denom^2 = A . (CC . A): lane w computes t_w = sum_w' CC[w][w'] A[w']
      float t0 = 0.f, t1 = 0.f;
      #pragma unroll 5
      for (int wp = 0; wp < 50; wp++)
        t0 += sCC[w0 * 50 + wp] * sAttn[wave][wp];
      if (w1 < 50) {
        #pragma unroll 5
        for (int wp = 0; wp < 50; wp++)
          t1 += sCC[w1 * 50 + wp] * sAttn[wave][wp];
      }
      float dsq = wredsum(A0 * t0 + A1 * t1);
      res = num / (sqrtf(dsq) + EPSF);
    }
    if (lane == 0) out[((size_t)i * 64 + c) * 36 + r] = res;
  }
}

// ============================================================================
extern "C" void kernel_launch(void* const* d_in, const int* in_sizes, int n_in,
                              void* d_out, int out_size, void* d_ws, size_t ws_size,
                              hipStream_t stream) {
  const float* imgs     = (const float*)d_in[0];   // [64,36,1024]
  const float* caps     = (const float*)d_in[1];   // [64,50,1024]
  const int*   img_lens = (const int*)d_in[2];     // [64]
  const int*   cap_lens = (const int*)d_in[3];     // [64]
  const float* alpha    = (const float*)d_in[4];   // scalar
  float* out = (float*)d_out;                      // [64,64,36]

  float* ws = (float*)d_ws;
  float* ni = ws;                        // 64*36
  float* nc = ws + 64 * 36;              // 64*50
  float* cc = ws + 64 * 36 + 64 * 50;    // 64*50*50

  const int nrows = 64 * 36 + 64 * 50;   // 5504 waves, 4 per block
  norms_kernel<<<dim3((nrows + 3) / 4), dim3(128), 0, stream>>>(imgs, caps, ni, nc);
  cc_kernel<<<dim3(64), dim3(256), 0, stream>>>(caps, cc);
  fused_kernel<<<dim3(64 * 64), dim3(192), 0, stream>>>(imgs, caps, img_lens,
                                                        cap_lens, alpha,
                                                        ni, nc, cc, out);
}